// PseudoMaskGenerator_58506044506691
// MI455X (gfx1250) — compile-verified
//
#include <hip/hip_runtime.h>
#include <hip/hip_bf16.h>
#include <stdint.h>

// ---------------------------------------------------------------------------
// Weighted K-means pseudo-mask generator for MI455X (gfx1250, wave32).
//  * Distances on the matrix path: V_WMMA_F32_16X16X4_F32 computes the
//    16(cluster) x 16(pixel) score tile  D = A(16x4) * B(4x16):
//      A row m = [-2*ch_m, -2*cw_m, |c_m|^2, 0]  (clusters, padded with +BIG)
//      B col n = [ h_n,     w_n,     1,      0]  (pixels)
//    argmin_k D[k][n] == argmin_k d^2(pixel n, cluster k)  (|p|^2 drops out).
//  * Lloyd iterations stream the mask through LDS with the Tensor Data Mover
//    (tensor_load_to_lds, TENSORcnt), double-buffered.
//  * Final 335 MB one-hot write uses non-temporal stores (output > L2).
// ---------------------------------------------------------------------------

typedef __attribute__((ext_vector_type(2))) float v2f;
typedef __attribute__((ext_vector_type(8))) float v8f;
typedef __attribute__((ext_vector_type(4))) unsigned int v4u;
typedef __attribute__((ext_vector_type(8))) int v8i_;
typedef __attribute__((ext_vector_type(4))) int v4i_;

#define KC    5
#define HH    512
#define WW    512
#define PTOT  (HH * WW)          // 262144 pixels per problem
#define ITERS 10
#define BIGF  3.0e38f
#define TILE  4096               // floats per TDM tile (16 KB)
#define NT    (PTOT / TILE)      // 64 tiles per problem

// ws layout (floats): [0 .. nProb)                  : n_fg per problem
//                     [64 + prob*16 + k*2 + {0,1})  : centroid (h,w)
#define WS_CENT_BASE 64

// ---------------------------------------------------------------------------
__global__ void pmg_init_ws(float* __restrict__ ws, int nFloats) {
  int i = blockIdx.x * blockDim.x + threadIdx.x;
  if (i < nFloats) ws[i] = 0.0f;
}

// ---------------------------------------------------------------------------
// Seed: per (problem, segment) argmax of w * uniform-hash score; also sums fg.
__global__ void pmg_seed(const float* __restrict__ mask, float* __restrict__ ws) {
  const int prob = blockIdx.x / KC;
  const int seg  = blockIdx.x % KC;
  const int tid  = threadIdx.x;
  const float* m = mask + (size_t)prob * PTOT;

  const int s0 = (int)(((long long)seg * PTOT) / KC);
  const int s1 = (int)(((long long)(seg + 1) * PTOT) / KC);

  float bestS = -1.0f;
  int   bestP = s0;
  float fg = 0.0f;
  for (int p = s0 + tid; p < s1; p += blockDim.x) {
    float w = m[p];
    __builtin_prefetch(m + p + blockDim.x, 0, 1);
    fg += w;
    uint32_t x = (uint32_t)p * 0x9E3779B9u + (uint32_t)prob * 0x85EBCA6Bu + 0x2545F491u;
    x ^= x >> 16; x *= 0x7FEB352Du;
    x ^= x >> 15; x *= 0x846CA68Bu;
    x ^= x >> 16;
    float u = (float)(x >> 8) * (1.0f / 16777216.0f);
    float s = w * u;
    if (s > bestS) { bestS = s; bestP = p; }
  }

  __shared__ float sS[256];
  __shared__ int   sP[256];
  __shared__ float sF[256];
  sS[tid] = bestS; sP[tid] = bestP; sF[tid] = fg;
  __syncthreads();
  for (int off = 128; off > 0; off >>= 1) {
    if (tid < off) {
      if (sS[tid + off] > sS[tid]) { sS[tid] = sS[tid + off]; sP[tid] = sP[tid + off]; }
      sF[tid] += sF[tid + off];
    }
    __syncthreads();
  }
  if (tid == 0) {
    int p = sP[0];
    ws[WS_CENT_BASE + prob * 16 + seg * 2 + 0] = (float)(p >> 9);
    ws[WS_CENT_BASE + prob * 16 + seg * 2 + 1] = (float)(p & (WW - 1));
    atomicAdd(&ws[prob], sF[0]);
  }
}

// ---------------------------------------------------------------------------
// Issue a 1-D TDM copy of nElem floats (nElem <= 65535) global -> LDS.
// D# group0: count=1 | lds_addr | global_addr[56:0] | type=2.
// D# group1: data_size=4B, tensor_dim0 = tile_dim0 = nElem, 1 row.
__device__ __forceinline__ void tdm_load_1d(uint32_t ldsOff, const float* gsrc,
                                            int nElem) {
  uint64_t ga = (uint64_t)(uintptr_t)gsrc;
  v4u g0;
  g0.x = 0x1u;                                        // count=1 (valid user D#)
  g0.y = ldsOff;                                      // lds_addr (bytes)
  g0.z = (uint32_t)ga;                                // global_addr[31:0]
  g0.w = (uint32_t)((ga >> 32) & 0x01FFFFFFu) | 0x80000000u; // [56:32] | type=2
  v8i_ g1;
  g1[0] = (int)(2u << 16);                            // data_size code 2 = 4B
  g1[1] = (int)((uint32_t)nElem << 16);               // tensor_dim0[15:0]
  g1[2] = (int)(1u << 16);                            // tensor_dim0 hi=0, dim1=1
  g1[3] = (int)((uint32_t)nElem << 16);               // tile_dim0
  g1[4] = 0;                                          // tile_dim1/2 = 0 (1-D)
  g1[5] = nElem;                                      // tensor_dim0_stride lo
  g1[6] = 0;
  g1[7] = 0;
  v4i_ z4 = {0, 0, 0, 0};
  v8i_ z8 = {0, 0, 0, 0, 0, 0, 0, 0};
  __builtin_amdgcn_tensor_load_to_lds(g0, g1, z4, z4, z8, 0);
}

__device__ __forceinline__ uint32_t lds_offset(const void* p) {
  return (uint32_t)(uintptr_t)p;   // shared-aperture generic ptr: low 32 = LDS off
}

// ---------------------------------------------------------------------------
// Build the per-lane A operand from centroids (cluster m = lane & 15).
__device__ __forceinline__ v2f make_A(const float* cent, int lane) {
  const int  mm = lane & 15;
  const bool lo = lane < 16;
  const bool ok = mm < KC;
  const int  ci = ok ? mm : 0;
  float ch = cent[ci * 2 + 0];
  float cw = cent[ci * 2 + 1];
  v2f a;
  a.x = lo ? (ok ? -2.0f * ch : 0.0f) : (ok ? (ch * ch + cw * cw) : BIGF);
  a.y = lo ? (ok ? -2.0f * cw : 0.0f) : 0.0f;
  return a;
}

// Two 16x16x4 f32 WMMAs cover a 32-pixel chunk; argmin cluster for pixels
// base+(lane&15) (k1) and base+16+(lane&15) (k2); valid in lanes 0-15.
__device__ __forceinline__ void assign32(v2f a, int base, int lane,
                                         int* k1o, int* k2o,
                                         float* h1o, float* x1o,
                                         float* h2o, float* x2o) {
  const int  mm = lane & 15;
  const bool lo = lane < 16;
  const int p1 = base + mm;
  const int p2 = p1 + 16;
  const float h1 = (float)(p1 >> 9), x1 = (float)(p1 & (WW - 1));
  const float h2 = (float)(p2 >> 9), x2 = (float)(p2 & (WW - 1));
  v2f b1, b2;
  b1.x = lo ? h1 : 1.0f;  b1.y = lo ? x1 : 0.0f;
  b2.x = lo ? h2 : 1.0f;  b2.y = lo ? x2 : 0.0f;
  v8f c1 = {}; v8f c2 = {};
  c1 = __builtin_amdgcn_wmma_f32_16x16x4_f32(false, a, false, b1, (short)0, c1, false, false);
  c2 = __builtin_amdgcn_wmma_f32_16x16x4_f32(false, a, false, b2, (short)0, c2, false, false);
  int k1 = 0; float bst = c1[0];
#pragma unroll
  for (int k = 1; k < KC; ++k) { if (c1[k] < bst) { bst = c1[k]; k1 = k; } }
  int k2 = 0; bst = c2[0];
#pragma unroll
  for (int k = 1; k < KC; ++k) { if (c2[k] < bst) { bst = c2[k]; k2 = k; } }
  *k1o = k1; *k2o = k2; *h1o = h1; *x1o = x1; *h2o = h2; *x2o = x2;
}

// ---------------------------------------------------------------------------
// 10 Lloyd iterations; one block (8 waves) per problem; centroids in LDS;
// mask streamed global->LDS by the Tensor Data Mover, double-buffered.
__global__ void pmg_iterate(const float* __restrict__ mask, float* __restrict__ ws) {
  const int prob   = blockIdx.x;
  const int tid    = threadIdx.x;
  const int lane   = tid & 31;
  const int waveId = tid >> 5;
  const int nWaves = blockDim.x >> 5;
  const int mm     = lane & 15;
  const bool lo    = lane < 16;
  const float* m   = mask + (size_t)prob * PTOT;

  __shared__ float sTile[2][TILE];   // 32 KB double buffer
  __shared__ float sCent[32];
  __shared__ float sAcc[16];
  if (tid < 32) sCent[tid] = 0.0f;
  __syncthreads();
  if (tid < 2 * KC) sCent[tid] = ws[WS_CENT_BASE + prob * 16 + tid];
  __syncthreads();

  for (int it = 0; it < ITERS; ++it) {
    v2f a = make_A(sCent, lane);
    float cnt[KC], sh[KC], sw[KC];
#pragma unroll
    for (int k = 0; k < KC; ++k) { cnt[k] = 0.0f; sh[k] = 0.0f; sw[k] = 0.0f; }

    if (waveId == 0) tdm_load_1d(lds_offset(&sTile[0][0]), m, TILE);

    for (int t = 0; t < NT; ++t) {
      const int cur = t & 1;
      if (waveId == 0) {
        if (t + 1 < NT) {     // prefetch next tile, then wait for current
          tdm_load_1d(lds_offset(&sTile[1 - cur][0]), m + (t + 1) * TILE, TILE);
          __builtin_amdgcn_s_wait_tensorcnt(1);  // in-order: current tile done
        } else {
          __builtin_amdgcn_s_wait_tensorcnt(0);
        }
      }
      __syncthreads();                           // tile[cur] visible to all

      for (int c = waveId; c < TILE / 32; c += nWaves) {
        const int idx  = c << 5;
        const int base = t * TILE + idx;
        float wv  = sTile[cur][idx + lane];      // weight of pixel base+lane
        float wv2 = sTile[cur][idx + 16 + mm];   // weight of partner pixel p2
        int k1, k2; float h1, x1, h2, x2;
        assign32(a, base, lane, &k1, &k2, &h1, &x1, &h2, &x2);
        if (lo) {
#pragma unroll
          for (int k = 0; k < KC; ++k) {
            float m1 = (k == k1) ? wv  : 0.0f;
            float m2 = (k == k2) ? wv2 : 0.0f;
            cnt[k] += m1 + m2;
            sh[k]  += m1 * h1 + m2 * h2;
            sw[k]  += m1 * x1 + m2 * x2;
          }
        }
      }
      __syncthreads();                 // all reads of tile[cur] done before DMA reuse
    }

    if (tid < 16) sAcc[tid] = 0.0f;
    __syncthreads();
    if (lo) {
#pragma unroll
      for (int k = 0; k < KC; ++k) {
        atomicAdd(&sAcc[k],          cnt[k]);
        atomicAdd(&sAcc[KC + k],     sh[k]);
        atomicAdd(&sAcc[2 * KC + k], sw[k]);
      }
    }
    __syncthreads();
    if (tid < KC) {                    // centroid update (keep old if empty)
      float c = sAcc[tid];
      if (c > 0.0f) {
        sCent[tid * 2 + 0] = sAcc[KC + tid]     / c;
        sCent[tid * 2 + 1] = sAcc[2 * KC + tid] / c;
      }
    }
    __syncthreads();
  }
  if (tid < 2 * KC) ws[WS_CENT_BASE + prob * 16 + tid] = sCent[tid];
}

// ---------------------------------------------------------------------------
// Final assignment + one-hot plane writes (the 335 MB bandwidth-bound part).
// Grid: nProb * 1024 blocks x 256 threads; each wave owns one 32-pixel chunk.
// Output exceeds L2 and is never re-read -> non-temporal stores.
__global__ void pmg_finalize(const float* __restrict__ mask,
                             const float* __restrict__ ws,
                             float* __restrict__ out) {
  const int prob   = blockIdx.x >> 10;
  const int bi     = blockIdx.x & 1023;
  const int tid    = threadIdx.x;
  const int lane   = tid & 31;
  const int waveId = tid >> 5;
  const int mm     = lane & 15;
  const bool lo    = lane < 16;
  const int base   = ((bi << 3) + waveId) << 5;        // chunk * 32
  const float* m   = mask + (size_t)prob * PTOT;

  float cent[2 * KC];
#pragma unroll
  for (int i = 0; i < 2 * KC; ++i) cent[i] = ws[WS_CENT_BASE + prob * 16 + i];
  const float nfg = ws[prob];

  v2f a = make_A(cent, lane);
  float wv = m[base + lane];
  int k1, k2; float h1, x1, h2, x2;
  assign32(a, base, lane, &k1, &k2, &h1, &x1, &h2, &x2);

  // Redistribute so every lane owns pixel (base + lane): lanes 16-31 pull k2
  // from lane-16; all 32 lanes then store fully coalesced plane writes.
  int kOther = __shfl(k2, mm, 32);
  int kMine  = lo ? k1 : kOther;

  const bool trivial = !(nfg > (float)KC);
  size_t o = (size_t)prob * KC * PTOT + (size_t)(base + lane);
#pragma unroll
  for (int k = 0; k < KC; ++k) {
    bool hit = trivial ? (k == 0) : (kMine == k);
    __builtin_nontemporal_store(hit ? wv : 0.0f, &out[o + (size_t)k * PTOT]);
  }
}

// ---------------------------------------------------------------------------
extern "C" void kernel_launch(void* const* d_in, const int* in_sizes, int n_in,
                              void* d_out, int out_size, void* d_ws, size_t ws_size,
                              hipStream_t stream) {
  const float* mask = (const float*)d_in[0];
  float* out = (float*)d_out;
  float* ws  = (float*)d_ws;

  const int nProb = in_sizes[0] / PTOT;                // B*C = 64
  const int wsFloats = WS_CENT_BASE + nProb * 16;

  pmg_init_ws<<<(wsFloats + 255) / 256, 256, 0, stream>>>(ws, wsFloats);
  pmg_seed<<<nProb * KC, 256, 0, stream>>>(mask, ws);
  pmg_iterate<<<nProb, 256, 0, stream>>>(mask, ws);
  pmg_finalize<<<nProb * 1024, 256, 0, stream>>>(mask, ws, out);
}